// LIFLayer_4664334483975
// MI455X (gfx1250) — compile-verified
//
#include <hip/hip_runtime.h>
#include <cstdint>

// Problem geometry (matches reference): T=512, B=16, N=8192
#define LIF_T        512
#define LIF_BN       (16 * 8192)          // floats per timestep = 131072
#define VEC_PER_T    (LIF_BN / 4)         // float4 per timestep  = 32768
#define BLOCK_THREADS 128
#define GRID_BLOCKS  (VEC_PER_T / BLOCK_THREADS)   // 256 blocks
#define DEPTH        32                    // pipeline stages (power of 2)
#define STAGE_BYTES  1024                  // per wave: 512B input + 512B noise
#define WAVE_LDS     (DEPTH * STAGE_BYTES) // 32 KB per wave
#define BLOCK_WAVES  (BLOCK_THREADS / 32)
#define SMEM_BYTES   (BLOCK_WAVES * WAVE_LDS)      // 128 KB per block (<320KB/WGP)

typedef float v4f __attribute__((ext_vector_type(4)));

// One LIF step for a single neuron; V and Vth carried in registers.
__device__ __forceinline__ float lif_step(float& V, float& Vth, float I, float e) {
    V = fmaf(0.05f, I - V, V);        // V += dt/tau * (I - V), dt/tau = 1/20
    V = fmaf(0.1f, e, V);             // + noise_std * eps
    const bool  sp = (V >= Vth);
    const float s  = sp ? 1.0f : 0.0f;
    V   = sp ? 0.0f : V;                              // reset
    Vth = fminf(fmaxf(fmaf(0.1f, s, Vth), 0.5f), 2.0f); // adaptive clip
    return s;
}

// ext_vector elements cannot bind to float&; stage through locals.
__device__ __forceinline__ v4f lif_step4(v4f& V, v4f& Vth, const v4f I, const v4f e) {
    v4f s;
    {
        float v = V.x, th = Vth.x;
        s.x = lif_step(v, th, I.x, e.x);
        V.x = v; Vth.x = th;
    }
    {
        float v = V.y, th = Vth.y;
        s.y = lif_step(v, th, I.y, e.y);
        V.y = v; Vth.y = th;
    }
    {
        float v = V.z, th = Vth.z;
        s.z = lif_step(v, th, I.z, e.z);
        V.z = v; Vth.z = th;
    }
    {
        float v = V.w, th = Vth.w;
        s.w = lif_step(v, th, I.w, e.w);
        V.w = v; Vth.w = th;
    }
    return s;
}

// CDNA5 async copy: 16B per lane, global -> LDS, tracked by ASYNCcnt.
// Streamed-once data: non-temporal so the 512 MB read stream skips L2 allocation.
__device__ __forceinline__ void async_load_b128(uint32_t lds_addr, const void* gaddr) {
    asm volatile("global_load_async_to_lds_b128 %0, %1, off th:TH_LOAD_NT"
                 :: "v"(lds_addr), "v"((uint64_t)gaddr)
                 : "memory");
}

__global__ __launch_bounds__(BLOCK_THREADS) void
lif_async_kernel(const float* __restrict__ input,
                 const float* __restrict__ noise,
                 float* __restrict__ out)
{
    __shared__ __align__(16) unsigned char smem[SMEM_BYTES];

    const int    tid  = threadIdx.x;
    const int    lane = tid & 31;
    const int    wave = tid >> 5;
    const size_t g    = (size_t)blockIdx.x * BLOCK_THREADS + tid; // float4 slot

    // Hardware LDS byte address for the async instruction (low 32 bits of the
    // flat shared-aperture address are the LDS offset).
    const uint32_t lds_abs   = (uint32_t)(uintptr_t)&smem[0];
    const uint32_t wave_base = lds_abs + (uint32_t)(wave * WAVE_LDS + lane * 16);
    unsigned char* const wave_cpu = smem + wave * WAVE_LDS + lane * 16;

    const v4f* __restrict__ in4  = (const v4f*)input;
    const v4f* __restrict__ ns4  = (const v4f*)noise;
    v4f* __restrict__       out4 = (v4f*)out;

    v4f V   = {0.f, 0.f, 0.f, 0.f};
    v4f Vth = {1.f, 1.f, 1.f, 1.f};

    // ---- prologue: fill all DEPTH pipeline stages (2*DEPTH asyncs/wave;
    // ASYNCcnt issue-stalls benignly at its 63 ceiling) ----
    #pragma unroll
    for (int d = 0; d < DEPTH; ++d) {
        async_load_b128(wave_base + d * STAGE_BYTES,       in4 + (size_t)d * VEC_PER_T + g);
        async_load_b128(wave_base + d * STAGE_BYTES + 512, ns4 + (size_t)d * VEC_PER_T + g);
    }

    // ---- steady state: consume stage t, refill with timestep t+DEPTH ----
    #pragma unroll 1
    for (int t = 0; t < LIF_T - DEPTH; ++t) {
        const int sidx = t & (DEPTH - 1);
        // Stage t's two loads are the oldest outstanding; done when
        // ASYNCcnt <= 2*(DEPTH-1) = 62 (async loads complete in order).
        asm volatile("s_wait_asynccnt 62" ::: "memory");
        const v4f I = *(const v4f*)(wave_cpu + sidx * STAGE_BYTES);
        const v4f E = *(const v4f*)(wave_cpu + sidx * STAGE_BYTES + 512);
        const v4f S = lif_step4(V, Vth, I, E);
        __builtin_nontemporal_store(S, &out4[(size_t)t * VEC_PER_T + g]);
        // Ensure this stage's LDS reads retired before overwriting the slot.
        asm volatile("s_wait_dscnt 0" ::: "memory");
        const size_t tp = (size_t)(t + DEPTH);
        async_load_b128(wave_base + sidx * STAGE_BYTES,       in4 + tp * VEC_PER_T + g);
        async_load_b128(wave_base + sidx * STAGE_BYTES + 512, ns4 + tp * VEC_PER_T + g);
    }

    // ---- epilogue: drain remaining DEPTH stages ----
    asm volatile("s_wait_asynccnt 0" ::: "memory");
    #pragma unroll 1
    for (int t = LIF_T - DEPTH; t < LIF_T; ++t) {
        const int sidx = t & (DEPTH - 1);
        const v4f I = *(const v4f*)(wave_cpu + sidx * STAGE_BYTES);
        const v4f E = *(const v4f*)(wave_cpu + sidx * STAGE_BYTES + 512);
        const v4f S = lif_step4(V, Vth, I, E);
        __builtin_nontemporal_store(S, &out4[(size_t)t * VEC_PER_T + g]);
    }
}

extern "C" void kernel_launch(void* const* d_in, const int* in_sizes, int n_in,
                              void* d_out, int out_size, void* d_ws, size_t ws_size,
                              hipStream_t stream) {
    const float* input = (const float*)d_in[0]; // (T,B,N) float32
    const float* noise = (const float*)d_in[1]; // (T,B,N) float32
    float*       out   = (float*)d_out;         // (T,B,N) float32 spikes
    (void)in_sizes; (void)n_in; (void)out_size; (void)d_ws; (void)ws_size;
    lif_async_kernel<<<GRID_BLOCKS, BLOCK_THREADS, 0, stream>>>(input, noise, out);
}